// NeuralPoints_34419867910571
// MI455X (gfx1250) — compile-verified
//
#include <hip/hip_runtime.h>
#include <cstdint>

// ---------------------------------------------------------------------------
// PIN-SLAM style neural-point map update check.
//   dist2[i], delta_travel[i], update_mask[i]  for i in [0, N)
// Outputs concatenated into d_out (float): [dist2 | delta_travel | mask(0/1)]
//
// Strategy (MI455X, 23.3 TB/s HBM, 192 MB L2):
//   - 40MB seg-min table + 40MB hash buffer + 24MB neural points are all
//     L2-resident -> random atomics/gathers run at L2 speed.
//   - pass1: hash + d_center, atomicMin(u32 bits of non-negative float),
//            stash {hw,d} packed in one u64 in ws, prefetch buffer_pt_index[hw].
//   - pass2: gathers + mask, NT stores for streamed outputs.
//   - BUFFER_SIZE is a reference module constant (like RESOLUTION/PRIMES):
//     hard-coded so the i64 modulo is a magic-multiply, not a software divide.
//   - grid coords computed in i32 (values ~ +-1000) -> v_mad_i64_i32 for the
//     hash products instead of full 64-bit multiplies.
// ---------------------------------------------------------------------------

#define RESOLUTION   0.3f
#define DIST2_THRESH 0.27f          // 3.0 * RESOLUTION^2 rounded to f32
#define DIFF_TRAVEL  120.0f         // 80.0 * 1.5
#define BUF_CONST    10000000LL     // reference BUFFER_SIZE

__global__ __launch_bounds__(256) void np_init_segmin(unsigned* __restrict__ seg, int buf) {
    int t = blockIdx.x * blockDim.x + threadIdx.x;
    int base = t * 4;
    const unsigned FMAX_BITS = 0x7F7FFFFFu;   // FLT_MAX as bits
    if (base + 3 < buf) {
        uint4 v = make_uint4(FMAX_BITS, FMAX_BITS, FMAX_BITS, FMAX_BITS);
        *reinterpret_cast<uint4*>(seg + base) = v;
    } else {
        for (int k = base; k < buf; ++k) seg[k] = FMAX_BITS;
    }
}

__global__ __launch_bounds__(256) void np_pass1(
    const float* __restrict__ pts,          // [N,3]
    const int*   __restrict__ buf_idx,      // [BUF]
    unsigned*    __restrict__ seg,          // [BUF] float-bits seg-min
    unsigned long long* __restrict__ ws_hd, // [N] {d_bits:32 | hw:32}
    int n)
{
#pragma clang fp contract(off)
    int i = blockIdx.x * blockDim.x + threadIdx.x;
    if (i >= n) return;

    float px = pts[3 * i + 0];
    float py = pts[3 * i + 1];
    float pz = pts[3 * i + 2];

    // grid = floor(p / RES); values are tiny (~+-1000) so i32 is exact.
    int gx = (int)floorf(px / RESOLUTION);
    int gy = (int)floorf(py / RESOLUTION);
    int gz = (int)floorf(pz / RESOLUTION);

    // prime spatial hash in int64 (i32 x i32 -> i64 products).
    // jnp.fmod == C '%' for integers (sign of dividend); constant divisor
    // -> signed magic-multiply, no software division.
    long long s = (long long)gx * 73856093LL
                + (long long)gy * 19349669LL
                + (long long)gz * 83492791LL;
    long long h = s % BUF_CONST;
    if (h < 0) h += BUF_CONST;
    unsigned hw = (unsigned)h;

    // distance^2 to voxel center, same association as jnp: (dx2+dy2)+dz2, no FMA
    float cx = ((float)gx + 0.5f) * RESOLUTION;
    float cy = ((float)gy + 0.5f) * RESOLUTION;
    float cz = ((float)gz + 0.5f) * RESOLUTION;
    float dx = px - cx;
    float dy = py - cy;
    float dz = pz - cz;
    float d = (dx * dx + dy * dy) + dz * dz;
    unsigned dbits = __float_as_uint(d);

    // d >= 0 -> IEEE float order == unsigned bit order; min is order-invariant
    atomicMin(&seg[hw], dbits);

    ws_hd[i] = ((unsigned long long)dbits << 32) | (unsigned long long)hw;

    // warm L2 for pass2's random gather of the hash buffer (fits in 192MB L2)
    __builtin_prefetch((const void*)&buf_idx[hw], 0, 1);
}

__global__ __launch_bounds__(256) void np_pass2(
    const float* __restrict__ pts,          // [N,3]
    const float* __restrict__ neural,       // [M,3]
    const int*   __restrict__ buf_idx,      // [BUF]
    const float* __restrict__ travel,       // [N_FRAMES]
    const int*   __restrict__ pts_ts,       // [M]
    const int*   __restrict__ cur_ts_p,     // [1]
    const unsigned* __restrict__ seg,       // [BUF]
    const unsigned long long* __restrict__ ws_hd,  // [N] {d_bits:32 | hw:32}
    float* __restrict__ out,                // [3*N]
    int n, int m)
{
#pragma clang fp contract(off)
    int i = blockIdx.x * blockDim.x + threadIdx.x;
    if (i >= n) return;

    unsigned long long hd = __builtin_nontemporal_load(&ws_hd[i]);  // read-once
    unsigned hw = (unsigned)(hd & 0xFFFFFFFFull);
    float    d  = __uint_as_float((unsigned)(hd >> 32));

    // keep: this point achieved the per-voxel minimum (exact compare, same d)
    bool keep = (d <= __uint_as_float(seg[hw]));

    int hidx = buf_idx[hw];                       // -1 = empty slot
    int idxw = (hidx < 0) ? (hidx + m) : hidx;    // torch wrap of -1 -> M-1

    float px = pts[3 * i + 0];
    float py = pts[3 * i + 1];
    float pz = pts[3 * i + 2];

    float vx = neural[3 * idxw + 0] - px;
    float vy = neural[3 * idxw + 1] - py;
    float vz = neural[3 * idxw + 2] - pz;
    float dist2 = (vx * vx + vy * vy) + vz * vz;

    int ts  = pts_ts[idxw];
    int cur = *cur_ts_p;
    float dt = travel[cur] - travel[ts];

    bool upd = keep && ((hidx == -1) || (dist2 > DIST2_THRESH) || (dt > DIFF_TRAVEL));

    // streamed outputs: NT stores so they don't evict the L2-resident tables
    __builtin_nontemporal_store(dist2, &out[i]);
    __builtin_nontemporal_store(dt, &out[n + i]);
    __builtin_nontemporal_store(upd ? 1.0f : 0.0f, &out[2 * n + i]);
}

extern "C" void kernel_launch(void* const* d_in, const int* in_sizes, int n_in,
                              void* d_out, int out_size, void* d_ws, size_t ws_size,
                              hipStream_t stream) {
    const float* pts     = (const float*)d_in[0];   // [N,3] f32
    const float* neural  = (const float*)d_in[1];   // [M,3] f32
    const int*   bufidx  = (const int*)d_in[2];     // [BUF] (values fit i32)
    const float* travel  = (const float*)d_in[3];   // [N_FRAMES] f32
    const int*   ptsts   = (const int*)d_in[4];     // [M]
    const int*   curts   = (const int*)d_in[5];     // [1]

    int n   = in_sizes[0] / 3;
    int m   = in_sizes[1] / 3;
    int buf = in_sizes[2];   // == BUF_CONST for the reference

    // workspace layout: [seg-min bits: buf u32][{hw,d}: n u64]
    unsigned char* ws = (unsigned char*)d_ws;
    unsigned* seg = (unsigned*)ws;
    unsigned long long* ws_hd = (unsigned long long*)(ws + (size_t)buf * 4);
    float* out = (float*)d_out;

    const int BLK = 256;
    int init_threads = (buf + 3) / 4;
    np_init_segmin<<<(init_threads + BLK - 1) / BLK, BLK, 0, stream>>>(seg, buf);
    np_pass1<<<(n + BLK - 1) / BLK, BLK, 0, stream>>>(pts, bufidx, seg, ws_hd, n);
    np_pass2<<<(n + BLK - 1) / BLK, BLK, 0, stream>>>(
        pts, neural, bufidx, travel, ptsts, curts, seg, ws_hd, out, n, m);
}